// GraspLoss_23210003268387
// MI455X (gfx1250) — compile-verified
//
#include <hip/hip_runtime.h>
#include <hip/hip_bf16.h>

#define Bn 32
#define Hn 512
#define Wn 512
#define HWn (Hn * Wn)

typedef __attribute__((ext_vector_type(2))) float v2f;
typedef __attribute__((ext_vector_type(4))) float v4f;
typedef __attribute__((ext_vector_type(8))) float v8f;

__device__ __forceinline__ float sl1(float x) {
    float ax = fabsf(x);
    return ax < 1.0f ? 0.5f * x * x : ax - 0.5f;
}

// ---------------- init: d_ws is poisoned, zero accumulators -----------------
__global__ __launch_bounds__(64) void grasp_init_ws(float* __restrict__ wsf,
                                                    unsigned long long* __restrict__ amax) {
    int t = threadIdx.x;
    if (t < 5) wsf[t] = 0.0f;
    if (t < 32) amax[t] = 0ull;
}

// ---------------- main streaming reduction ----------------------------------
// grid = (HW/1024, B), block = 256. Each thread: one float4 per plane (NT loads).
__global__ __launch_bounds__(256) void grasp_reduce_main(
    const float* __restrict__ pq, const float* __restrict__ pa,
    const float* __restrict__ pw, const float* __restrict__ pt,
    const float* __restrict__ gq, const float* __restrict__ ga,
    const float* __restrict__ gw, const float* __restrict__ gt,
    const float* __restrict__ am,
    float* __restrict__ wsf, unsigned long long* __restrict__ amax)
{
    const int b    = blockIdx.y;
    const int tid  = threadIdx.x;
    const int p0   = (blockIdx.x * 256 + tid) * 4;   // pixel index within plane
    const int base = b * HWn + p0;                   // [B,1,H,W] flat
    const int ab0  = b * 2 * HWn + p0;               // [B,2,H,W] chan 0
    const int ab1  = ab0 + HWn;                      // chan 1

    // Non-temporal b128 loads: data read exactly once, working set > L2.
    v4f vpq  = __builtin_nontemporal_load((const v4f*)(pq + base));
    v4f vgq  = __builtin_nontemporal_load((const v4f*)(gq + base));
    v4f vpw  = __builtin_nontemporal_load((const v4f*)(pw + base));
    v4f vgw  = __builtin_nontemporal_load((const v4f*)(gw + base));
    v4f vpt  = __builtin_nontemporal_load((const v4f*)(pt + base));
    v4f vgt  = __builtin_nontemporal_load((const v4f*)(gt + base));
    v4f vm   = __builtin_nontemporal_load((const v4f*)(am + base));
    v4f vpa0 = __builtin_nontemporal_load((const v4f*)(pa + ab0));
    v4f vpa1 = __builtin_nontemporal_load((const v4f*)(pa + ab1));
    v4f vga0 = __builtin_nontemporal_load((const v4f*)(ga + ab0));
    v4f vga1 = __builtin_nontemporal_load((const v4f*)(ga + ab1));

    float q = 0.f, aa = 0.f, ww = 0.f, tt = 0.f, mm = 0.f;
    float bestv = -1.0f;
    unsigned besti = 0u;

#pragma unroll
    for (int j = 0; j < 4; ++j) {
        float gqv = vgq[j];
        float d   = vpq[j] - gqv;
        float wmp = (gqv > 0.1f) ? 6.0f : 1.0f;
        q += d * d * wmp;
        float mj = vm[j];
        mm += mj;
        aa += sl1((vpa0[j] - vga0[j]) * mj) + sl1((vpa1[j] - vga1[j]) * mj);
        ww += sl1((vpw[j] - vgw[j]) * mj);
        tt += sl1((vpt[j] - vgt[j]) * mj);
        if (gqv > bestv) { bestv = gqv; besti = (unsigned)(p0 + j); } // first-max per thread
    }

    // wave32 reductions (sum x5, argmax with first-index tie-break)
    unsigned vbits = __float_as_uint(bestv);   // bestv >= 0 -> monotonic bit order
    unsigned ibits = besti;
    for (int off = 16; off > 0; off >>= 1) {
        q  += __shfl_down(q,  off, 32);
        aa += __shfl_down(aa, off, 32);
        ww += __shfl_down(ww, off, 32);
        tt += __shfl_down(tt, off, 32);
        mm += __shfl_down(mm, off, 32);
        unsigned ov = __shfl_down(vbits, off, 32);
        unsigned oi = __shfl_down(ibits, off, 32);
        if (ov > vbits || (ov == vbits && oi < ibits)) { vbits = ov; ibits = oi; }
    }

    __shared__ float    part[5][8];
    __shared__ unsigned avb[8], aib[8];
    int wid = tid >> 5, lane = tid & 31;
    if (lane == 0) {
        part[0][wid] = q;  part[1][wid] = aa; part[2][wid] = ww;
        part[3][wid] = tt; part[4][wid] = mm;
        avb[wid] = vbits;  aib[wid] = ibits;
    }
    __syncthreads();

    if (tid < 32) {   // wave 0: EXEC all-1s (required for WMMA)
        // Fold 8 wave-partials x 5 quantities with one v_wmma_f32_16x16x4_f32:
        // A[m][k] = part[m][k] + part[m][k+4] (rows m=0..4 hold quantities),
        // B = ones(4x16)  =>  D[m][n] = total[m] replicated over n.
        int m  = tid & 15;
        int kb = (tid < 16) ? 0 : 2;
        v2f A = {0.f, 0.f};
        if (m < 5) {
            A.x = part[m][kb]     + part[m][kb + 4];
            A.y = part[m][kb + 1] + part[m][kb + 5];
        }
        v2f Bv = {1.f, 1.f};
        v8f C  = {};
#if defined(__gfx1250__) && __has_builtin(__builtin_amdgcn_wmma_f32_16x16x4_f32)
        C = __builtin_amdgcn_wmma_f32_16x16x4_f32(false, A, false, Bv,
                                                  (short)0, C, false, false);
        if (tid == 0) {
            atomicAdd(&wsf[0], C[0]); atomicAdd(&wsf[1], C[1]);
            atomicAdd(&wsf[2], C[2]); atomicAdd(&wsf[3], C[3]);
            atomicAdd(&wsf[4], C[4]);
        }
#else
        if (tid < 5) {
            float s = 0.f;
            for (int w = 0; w < 8; ++w) s += part[tid][w];
            atomicAdd(&wsf[tid], s);
        }
#endif
        if (tid == 0) {
            unsigned bv = avb[0], bi = aib[0];
            for (int w = 1; w < 8; ++w)
                if (avb[w] > bv || (avb[w] == bv && aib[w] < bi)) { bv = avb[w]; bi = aib[w]; }
            unsigned long long key =
                ((unsigned long long)bv << 32) | (unsigned long long)(~bi);
            atomicMax(&amax[b], key);
        }
    }
}

// ---------------- endpoint-loss tail + final combine (one wave32) -----------
__global__ __launch_bounds__(32) void grasp_finalize(
    const float* __restrict__ pa, const float* __restrict__ pw,
    const float* __restrict__ ga, const float* __restrict__ gw,
    const float* __restrict__ wsf, const unsigned long long* __restrict__ amax,
    float* __restrict__ out)
{
    int b = threadIdx.x;                      // 0..31 == batch index
    unsigned long long key = amax[b];
    unsigned idx = ~(unsigned)(key & 0xFFFFFFFFull);   // flat argmax within plane
    int iy = (int)(idx / Wn), ix = (int)(idx % Wn);

    int ab0 = b * 2 * HWn + (int)idx;
    int ab1 = ab0 + HWn;
    int wb  = b * HWn + (int)idx;

    float g_c = ga[ab0], g_s = ga[ab1];
    float p_c = pa[ab0], p_s = pa[ab1];
    float g_th = 0.5f * atan2f(g_s, g_c);
    float p_th = 0.5f * atan2f(p_s, p_c);
    const float scale = 512.0f;
    float g_half = 0.5f * gw[wb] * scale;
    float p_half = 0.5f * pw[wb] * scale;
    float gux = g_half * cosf(g_th), guy = g_half * sinf(g_th);
    float pux = p_half * cosf(p_th), puy = p_half * sinf(p_th);
    (void)iy; (void)ix;  // centers cancel: (gx - pux) - (gx - gux) = gux - pux

    float endb = 0.25f * (sl1(gux - pux) + sl1(guy - puy) +
                          sl1(pux - gux) + sl1(puy - guy));

    for (int off = 16; off > 0; off >>= 1)
        endb += __shfl_down(endb, off, 32);

    if (b == 0) {
        float msum = wsf[4];
        float ql = wsf[0] / (float)(Bn * HWn);
        float al = wsf[1] / (2.0f * msum + 1e-6f);
        float wl = wsf[2] / (msum + 1e-6f);
        float tl = wsf[3] / (msum + 1e-6f);
        float el = endb * (1.0f / (float)Bn);
        out[0] = 10.0f * ql + 1.0f * al + 1.0f * wl + 1.0f * tl + 0.3f * el;
    }
}

extern "C" void kernel_launch(void* const* d_in, const int* in_sizes, int n_in,
                              void* d_out, int out_size, void* d_ws, size_t ws_size,
                              hipStream_t stream) {
    (void)in_sizes; (void)n_in; (void)out_size; (void)ws_size;
    const float* pq = (const float*)d_in[0];
    const float* pa = (const float*)d_in[1];
    const float* pw = (const float*)d_in[2];
    const float* pt = (const float*)d_in[3];
    const float* gq = (const float*)d_in[4];
    const float* ga = (const float*)d_in[5];
    const float* gw = (const float*)d_in[6];
    const float* gt = (const float*)d_in[7];
    const float* am = (const float*)d_in[8];

    float* wsf = (float*)d_ws;                                        // 5 floats
    unsigned long long* amax = (unsigned long long*)((char*)d_ws + 64); // 32 u64

    grasp_init_ws<<<1, 64, 0, stream>>>(wsf, amax);

    dim3 grid(HWn / (256 * 4), Bn);   // (256, 32)
    grasp_reduce_main<<<grid, 256, 0, stream>>>(pq, pa, pw, pt, gq, ga, gw, gt, am,
                                                wsf, amax);

    grasp_finalize<<<1, 32, 0, stream>>>(pa, pw, ga, gw, wsf, amax, (float*)d_out);
}